// ParametricInterpolation_5961414606876
// MI455X (gfx1250) — compile-verified
//
#include <hip/hip_runtime.h>

// ParametricInterpolation: out[b,i] = x[b,np]*(1-k) + x[b,np-1]*k
// where curve = quartic(params[b]/scaler, i), k = frac(curve) (RNE),
// np = clip(i - rint(curve), 1, S-1).
//
// Memory-bound (~256 MB @ 23.3 TB/s => ~11 us floor). Strategy:
//  - 1 workgroup per row, 256 threads (8 waves, wave32).
//  - Async-copy the whole 8 KB row global->LDS (ASYNCcnt path,
//    global_load_async_to_lds_b128 + s_wait_asynccnt), so the
//    data-dependent gather is served by LDS ds_load_b32 instead of
//    uncoalesced VMEM, and each x element is fetched from HBM once.
//  - Coalesced b128 staging loads, coalesced b32 output stores.

#define S_LEN 2048
#define BLOCK 256
#define PER_THREAD (S_LEN / BLOCK) // 8

// Builtin expects: (int4 addrspace(1)*, int4 addrspace(3)*, imm, imm)
// per the hipcc diagnostic ("__vector_size__(4*sizeof(int)) int __device__ *").
typedef int v4i __attribute__((vector_size(16)));
typedef __attribute__((address_space(1))) v4i* gv4i_ptr;  // global
typedef __attribute__((address_space(3))) v4i* lv4i_ptr;  // LDS

__global__ __launch_bounds__(BLOCK) void
ParametricInterpolation_kernel(const float* __restrict__ x,
                               const float* __restrict__ params,
                               const float* __restrict__ scaler,
                               float* __restrict__ out)
{
    __shared__ __align__(16) float row[S_LEN];

    const int b = blockIdx.x;
    const int t = threadIdx.x;
    const float* __restrict__ xrow = x + (size_t)b * S_LEN;
    float* __restrict__ orow       = out + (size_t)b * S_LEN;

    // ---- Stage row into LDS: 512 x float4, 2 per thread ----
    {
        const float4* g = (const float4*)xrow;  // row base is 8KB-aligned
        float4*       l = (float4*)row;
#if __has_builtin(__builtin_amdgcn_global_load_async_to_lds_b128)
        // gfx1250 async global->LDS (tracked by ASYNCcnt)
        __builtin_amdgcn_global_load_async_to_lds_b128(
            (gv4i_ptr)(g + t),         (lv4i_ptr)(l + t),         0, 0);
        __builtin_amdgcn_global_load_async_to_lds_b128(
            (gv4i_ptr)(g + t + BLOCK), (lv4i_ptr)(l + t + BLOCK), 0, 0);
  #if __has_builtin(__builtin_amdgcn_s_wait_asynccnt)
        __builtin_amdgcn_s_wait_asynccnt(0);   // drain this wave's async copies
  #endif
#else
        l[t]         = g[t];
        l[t + BLOCK] = g[t + BLOCK];
#endif
    }

    // ---- Per-row coefficients (uniform broadcast loads -> s_load) ----
    const float p0 = params[(size_t)b * 5 + 0] / scaler[0];
    const float p1 = params[(size_t)b * 5 + 1] / scaler[1];
    const float p2 = params[(size_t)b * 5 + 2] / scaler[2];
    const float p3 = params[(size_t)b * 5 + 3] / scaler[3];
    const float p4 = params[(size_t)b * 5 + 4] / scaler[4];

    __syncthreads();   // all waves' async data now visible in LDS

    const float fS1 = (float)(S_LEN - 1);
#pragma unroll
    for (int j = 0; j < PER_THREAD; ++j) {
        const int   i  = t + j * BLOCK;
        const float fi = (float)i;
        // Horner quartic
        float c  = fmaf(fmaf(fmaf(fmaf(p0, fi, p1), fi, p2), fi, p3), fi, p4);
        float ci = rintf(c);          // round-half-to-even == jnp.round
        float k  = c - ci;
        // np = clip(i - ci, 1, S-1); ci is integral so (fi - ci) is integral
        float posf = fminf(fmaxf(fi - ci, 1.0f), fS1);
        int   np   = (int)posf;
        float x1 = row[np];           // ds_load_b32 (LDS-served gather)
        float x2 = row[np - 1];
        orow[i] = fmaf(1.0f - k, x1, k * x2);   // x1*(1-k) + x2*k
    }
}

extern "C" void kernel_launch(void* const* d_in, const int* in_sizes, int n_in,
                              void* d_out, int out_size, void* d_ws, size_t ws_size,
                              hipStream_t stream)
{
    const float* x      = (const float*)d_in[0];
    const float* params = (const float*)d_in[1];
    const float* scaler = (const float*)d_in[2];
    float*       out    = (float*)d_out;

    const int B = in_sizes[0] / S_LEN;   // 16384 rows
    ParametricInterpolation_kernel<<<B, BLOCK, 0, stream>>>(x, params, scaler, out);
}